// CCM_77043123355820
// MI455X (gfx1250) — compile-verified
//
#include <hip/hip_runtime.h>

typedef float v2f __attribute__((ext_vector_type(2)));
typedef float v8f __attribute__((ext_vector_type(8)));

#define F_BINS 257

// Select element v (0..7) of a v8f accumulator without scratch.
__device__ __forceinline__ float sel8(v8f c, int v) {
    float r = c[0];
    r = (v == 1) ? c[1] : r;
    r = (v == 2) ? c[2] : r;
    r = (v == 3) ? c[3] : r;
    r = (v == 4) ? c[4] : r;
    r = (v == 5) ? c[5] : r;
    r = (v == 6) ? c[6] : r;
    r = (v == 7) ? c[7] : r;
    return r;
}

// One wave32 per 16-frequency tile. The 16x16x4 F32 WMMA computes, via its
// diagonal, 16 per-frequency complex filter dot products (K=20, 5 chunks of 4).
__global__ __launch_bounds__(32)
void ccm_wmma_kernel(const float* __restrict__ m,
                     const float* __restrict__ x,
                     const float* __restrict__ cache,
                     float* __restrict__ out)
{
    const int  lane = threadIdx.x;      // 0..31
    const int  base = blockIdx.x * 16;  // frequency tile base
    const int  n    = lane & 15;        // A-row == B-column index within tile
    const bool hi   = lane >= 16;       // upper lane half -> K block {2,3}
    const int  f    = base + n;         // this lane's frequency
    const bool fvalid = (f < F_BINS);

    v8f cr = {0.f, 0.f, 0.f, 0.f, 0.f, 0.f, 0.f, 0.f};  // real-part chain
    v8f ci = {0.f, 0.f, 0.f, 0.f, 0.f, 0.f, 0.f, 0.f};  // imag-part chain

#pragma unroll
    for (int c = 0; c < 5; ++c) {
        // K indices this lane supplies for chunk c:
        //   lanes 0-15 : k = 4c+0 (vgpr0), 4c+1 (vgpr1)  -> tap pair q = 2c
        //   lanes 16-31: k = 4c+2 (vgpr0), 4c+3 (vgpr1)  -> tap pair q = 2c+1
        // Even k carries Mr (A) / Xr or Xi (B); odd k carries Mi / -Xi or Xr.
        const int q = 2 * c + (hi ? 1 : 0);   // tap index 0..8 (9 = zero pad)

        float Mr = 0.f, Mi = 0.f, Xr = 0.f, Xi = 0.f;
        if (q < 9 && fvalid) {
            // A coefficients: complex taps from m via fixed 2x3 projection _V
            const float m0 = m[(q     ) * F_BINS + f];
            const float m1 = m[(q +  9) * F_BINS + f];
            const float m2 = m[(q + 18) * F_BINS + f];
            Mr = m0 - 0.5f * (m1 + m2);
            Mi = 0.86602540378443864676f * (m1 - m2);

            // B coefficients: signal xc[t] at freq f + j - 1 (zero padded)
            const int t = q / 3;
            const int j = q - 3 * t;
            const int g = f + j - 1;
            if (g >= 0 && g < F_BINS) {
                if (t < 2) {                    // frames 0,1 come from cache
                    Xr = cache[(g * 2 + t) * 2 + 0];
                    Xi = cache[(g * 2 + t) * 2 + 1];
                } else {                        // frame 2 is the new input x
                    Xr = x[g * 2 + 0];
                    Xi = x[g * 2 + 1];
                }
            }
        }

        const v2f A  = {Mr, Mi};
        const v2f Br = {Xr, -Xi};   // out_real = sum Mr*Xr + Mi*(-Xi)
        const v2f Bi = {Xi,  Xr};   // out_imag = sum Mr*Xi + Mi*( Xr)

        cr = __builtin_amdgcn_wmma_f32_16x16x4_f32(false, A, false, Br,
                                                   (short)0, cr, false, false);
        ci = __builtin_amdgcn_wmma_f32_16x16x4_f32(false, A, false, Bi,
                                                   (short)0, ci, false, false);
    }

    // Diagonal extraction: D[M==N] lives in lanes 0-7 (vgpr = lane) and
    // lanes 24-31 (vgpr = lane-24, N = lane-16).
    const bool isDiag = (lane < 8) || (lane >= 24);
    const int  v      = (lane < 8) ? lane : (lane - 24);
    if (isDiag && fvalid) {
        out[f * 2 + 0] = sel8(cr, v);
        out[f * 2 + 1] = sel8(ci, v);
    }

    // new_cache[f][0] = cache[f][1]; new_cache[f][1] = x[f]  (lanes 0-15 only)
    if (!hi && fvalid) {
        float* nc = out + 2 * F_BINS;   // x_enh occupies first 514 floats
        nc[(f * 2 + 0) * 2 + 0] = cache[(f * 2 + 1) * 2 + 0];
        nc[(f * 2 + 0) * 2 + 1] = cache[(f * 2 + 1) * 2 + 1];
        nc[(f * 2 + 1) * 2 + 0] = x[f * 2 + 0];
        nc[(f * 2 + 1) * 2 + 1] = x[f * 2 + 1];
    }
}

extern "C" void kernel_launch(void* const* d_in, const int* in_sizes, int n_in,
                              void* d_out, int out_size, void* d_ws, size_t ws_size,
                              hipStream_t stream) {
    (void)in_sizes; (void)n_in; (void)out_size; (void)d_ws; (void)ws_size;
    const float* m     = (const float*)d_in[0];
    const float* x     = (const float*)d_in[1];
    const float* cache = (const float*)d_in[2];
    float* out = (float*)d_out;

    // 17 tiles of 16 frequencies, one wave32 each: latency-bound problem,
    // spread across WGPs with minimal launch overhead.
    hipLaunchKernelGGL(ccm_wmma_kernel, dim3(17), dim3(32), 0, stream,
                       m, x, cache, out);
}